// Attention13_16939351016297
// MI455X (gfx1250) — compile-verified
//
#include <hip/hip_runtime.h>
#include <hip/hip_bf16.h>

typedef __attribute__((ext_vector_type(2))) float v2f;
typedef __attribute__((ext_vector_type(8))) float v8f;

// ---------------------------------------------------------------------------
// CDNA5 async global->LDS copy (ASYNCcnt-tracked), per ISA 15.18.3 op 98.
// lds generic pointer low 32 bits are the LDS byte address (aperture scheme).
// ---------------------------------------------------------------------------
__device__ __forceinline__ void async_cp16(void* lds, const void* g) {
    unsigned l32 = (unsigned)(unsigned long long)(uintptr_t)lds;
    asm volatile("global_load_async_to_lds_b128 %0, %1, off"
                 :: "v"(l32), "v"(g)
                 : "memory");
}
__device__ __forceinline__ void wait_async0() {
    asm volatile("s_wait_asynccnt 0" ::: "memory");
}

// ---------------------------------------------------------------------------
// Stage 1: per-batch correlation matrices via fp32 WMMA.
//   C_yx[b] = y[b] @ x[b]^T, C_yy[b] = y y^T, C_xx[b] = x x^T  (reduce over m)
// Grid: splitK(4) x [b*3+mat](12) x tile(8: 2 row-bands of 128 x 4 col-bands
// of 64) = 384 WGs, 256 thr. Deterministic split-K partial buffers.
// Each wave owns a 2x2 block of 16x16 subtiles (4 fragment loads -> 4 WMMAs).
// Ping-pong LDS + async copies overlap global traffic with WMMA.
// ---------------------------------------------------------------------------
__global__ __launch_bounds__(256) void corr_kernel(const float* __restrict__ x,
                                                   const float* __restrict__ y,
                                                   float* __restrict__ Cpart) {
    int bid  = blockIdx.x;
    int p    = bid & 3;          // split-K part
    int t2   = bid >> 2;
    int tile = t2 & 7;           // 2x4 grid of 128x64 tiles
    int t3   = t2 >> 3;          // 0..11 = b*3 + mat
    int mat  = t3 % 3;           // 0: yx, 1: yy, 2: xx
    int b    = t3 / 3;
    int ti = tile >> 2;          // 0..1  (128-row band)
    int tj = tile & 3;           // 0..3  (64-col band)

    const size_t M = 8192;
    const float* srcA = (mat == 2) ? x : y;   // C rows
    const float* srcB = (mat == 1) ? y : x;   // C cols
    const float* Ab = srcA + ((size_t)b * 256 + (size_t)ti * 128) * M;
    const float* Bb = srcB + ((size_t)b * 256 + (size_t)tj * 64) * M;
    float* Cb = Cpart + ((size_t)p * 12 + (size_t)t3) * 65536;

    __shared__ float ldsA[2][128][68];
    __shared__ float ldsB[2][64][68];

    int tid  = threadIdx.x;
    int wave = tid >> 5;
    int lane = tid & 31;
    int lrow = lane & 15;
    int khalf = (lane < 16) ? 0 : 2;

    // wave -> 2x2 block of subtiles: rows (si2*2 + {0,1}), cols (sj2*2 + {0,1})
    int si2 = wave >> 1;         // 0..3 over 8 row-subtiles
    int sj2 = wave & 1;          // 0..1 over 4 col-subtiles

    v8f acc00 = {}, acc01 = {}, acc10 = {}, acc11 = {};

    auto load_slab = [&](int buf, int mc) {
#pragma unroll
        for (int q = 0; q < 8; ++q) {            // A: 128x64 f32 = 2048 float4
            int idx = tid + q * 256;
            int row = idx >> 4;
            int c4  = (idx & 15) << 2;
            async_cp16(&ldsA[buf][row][c4], Ab + (size_t)row * M + mc + c4);
        }
#pragma unroll
        for (int q = 0; q < 4; ++q) {            // B: 64x64 f32 = 1024 float4
            int idx = tid + q * 256;
            int row = idx >> 4;
            int c4  = (idx & 15) << 2;
            async_cp16(&ldsB[buf][row][c4], Bb + (size_t)row * M + mc + c4);
        }
    };

    int mbeg = p * 2048, mend = mbeg + 2048;
    load_slab(0, mbeg);
    wait_async0();
    __syncthreads();

    int ra0 = (si2 * 2) * 16 + lrow;
    int ra1 = ra0 + 16;
    int rb0 = (sj2 * 2) * 16 + lrow;
    int rb1 = rb0 + 16;

    for (int mc = mbeg; mc < mend; mc += 64) {
        int cur = ((mc - mbeg) >> 6) & 1;
        if (mc + 64 < mend) load_slab(cur ^ 1, mc + 64);   // prefetch next slab
#pragma unroll
        for (int kk = 0; kk < 64; kk += 4) {
            v2f a0, a1, b0, b1;
            a0.x = ldsA[cur][ra0][kk + khalf];
            a0.y = ldsA[cur][ra0][kk + khalf + 1];
            a1.x = ldsA[cur][ra1][kk + khalf];
            a1.y = ldsA[cur][ra1][kk + khalf + 1];
            b0.x = ldsB[cur][rb0][kk + khalf];
            b0.y = ldsB[cur][rb0][kk + khalf + 1];
            b1.x = ldsB[cur][rb1][kk + khalf];
            b1.y = ldsB[cur][rb1][kk + khalf + 1];
            acc00 = __builtin_amdgcn_wmma_f32_16x16x4_f32(false, a0, false, b0,
                                                          (short)0, acc00, false, false);
            acc01 = __builtin_amdgcn_wmma_f32_16x16x4_f32(false, a0, false, b1,
                                                          (short)0, acc01, false, false);
            acc10 = __builtin_amdgcn_wmma_f32_16x16x4_f32(false, a1, false, b0,
                                                          (short)0, acc10, false, false);
            acc11 = __builtin_amdgcn_wmma_f32_16x16x4_f32(false, a1, false, b1,
                                                          (short)0, acc11, false, false);
        }
        wait_async0();
        __syncthreads();
    }

    // C/D layout: VGPR r -> row (r | r+8) in lane halves, col = lane%16
    int rowoff = (lane < 16) ? 0 : 8;
    v8f accs[2][2] = {{acc00, acc01}, {acc10, acc11}};
#pragma unroll
    for (int u = 0; u < 2; ++u)
#pragma unroll
        for (int v = 0; v < 2; ++v) {
            int si = si2 * 2 + u, sj = sj2 * 2 + v;
            float* Ct = Cb + (size_t)(ti * 128 + si * 16 + rowoff) * 256 +
                        tj * 64 + sj * 16 + lrow;
#pragma unroll
            for (int r = 0; r < 8; ++r) Ct[(size_t)r * 256] = accs[u][v][r];
        }
}

// ---------------------------------------------------------------------------
// Stage 2: reduce the 4 split-K partials.  N = 12 * 65536 elements.
// ---------------------------------------------------------------------------
__global__ __launch_bounds__(256) void reduce_kernel(const float* __restrict__ Cpart,
                                                     float* __restrict__ C) {
    const size_t N = 786432;  // 12 * 65536
    size_t i = ((size_t)blockIdx.x * 256 + threadIdx.x) * 4;
    float4 s0 = *(const float4*)(Cpart + i);
    float4 s1 = *(const float4*)(Cpart + N + i);
    float4 s2 = *(const float4*)(Cpart + 2 * N + i);
    float4 s3 = *(const float4*)(Cpart + 3 * N + i);
    float4 r;
    r.x = (s0.x + s1.x) + (s2.x + s3.x);
    r.y = (s0.y + s1.y) + (s2.y + s3.y);
    r.z = (s0.z + s1.z) + (s2.z + s3.z);
    r.w = (s0.w + s1.w) + (s2.w + s3.w);
    *(float4*)(C + i) = r;
}

// ---------------------------------------------------------------------------
// Stage 3: per-(b,h) attention weights (32x32) from correlation matrices.
//   G = w_q_h Cyx w_k_h^T ; norms from Cyy / Cxx diagonals; softmax rows.
// Grid: 32 WGs x 256 thr.   Call layout: [b][mat(3: yx,yy,xx)][256][256]
// ---------------------------------------------------------------------------
__global__ __launch_bounds__(256) void attn_kernel(const float* __restrict__ Call,
                                                   const float* __restrict__ w_q,
                                                   const float* __restrict__ w_kv,
                                                   float* __restrict__ attnOut) {
    int b = blockIdx.x >> 3, h = blockIdx.x & 7;
    const float* wq  = w_q  + (size_t)h * 32 * 256;   // [32][256]
    const float* wk  = w_kv + (size_t)h * 32 * 256;   // top half of w_kv = k-proj
    const float* cyx = Call + ((size_t)b * 3 + 0) * 65536;
    const float* cyy = Call + ((size_t)b * 3 + 1) * 65536;
    const float* cxx = Call + ((size_t)b * 3 + 2) * 65536;
    int tid = threadIdx.x;

    __shared__ float T[32][256];
    __shared__ float G[32][32];
    __shared__ float nrmq[32];
    __shared__ float nrmk[32];

    // T = w_q_h @ Cyx   (thread = column)
    for (int i = 0; i < 32; ++i) {
        float s = 0.f;
        for (int d = 0; d < 256; ++d) s += wq[i * 256 + d] * cyx[(size_t)d * 256 + tid];
        T[i][tid] = s;
    }
    __syncthreads();
    // G = T @ w_k_h^T
    for (int t = 0; t < 4; ++t) {
        int idx = t * 256 + tid;
        int i = idx >> 5, j = idx & 31;
        float s = 0.f;
        for (int c = 0; c < 256; ++c) s += T[i][c] * wk[j * 256 + c];
        G[i][j] = s;
    }
    __syncthreads();
    // ||q_i||^2 = diag(w_q_h Cyy w_q_h^T)
    for (int i = 0; i < 32; ++i) {
        float s = 0.f;
        for (int d = 0; d < 256; ++d) s += wq[i * 256 + d] * cyy[(size_t)d * 256 + tid];
        T[i][tid] = s;
    }
    __syncthreads();
    if (tid < 32) {
        float s = 0.f;
        for (int c = 0; c < 256; ++c) s += T[tid][c] * wq[tid * 256 + c];
        nrmq[tid] = sqrtf(fmaxf(s, 0.f));
    }
    __syncthreads();
    // ||k_j||^2 = diag(w_k_h Cxx w_k_h^T)
    for (int i = 0; i < 32; ++i) {
        float s = 0.f;
        for (int d = 0; d < 256; ++d) s += wk[i * 256 + d] * cxx[(size_t)d * 256 + tid];
        T[i][tid] = s;
    }
    __syncthreads();
    if (tid < 32) {
        float s = 0.f;
        for (int c = 0; c < 256; ++c) s += T[tid][c] * wk[tid * 256 + c];
        nrmk[tid] = sqrtf(fmaxf(s, 0.f));
    }
    __syncthreads();
    // scale + F.normalize(eps=1e-12) folding + stable row softmax
    if (tid < 32) {
        const int i = tid;
        const float scale = 1.0f / sqrtf(8192.0f);   // m^-0.5
        float qn = fmaxf(nrmq[i], 1e-12f);
        float row[32];
        float mx = -3.402823466e38f;
        for (int j = 0; j < 32; ++j) {
            float kn = fmaxf(nrmk[j], 1e-12f);
            float v = G[i][j] * scale / (qn * kn);
            row[j] = v;
            mx = fmaxf(mx, v);
        }
        float sum = 0.f;
        for (int j = 0; j < 32; ++j) { row[j] = __expf(row[j] - mx); sum += row[j]; }
        float inv = 1.f / sum;
        float* ao = attnOut + ((size_t)(b * 8 + h)) * 1024 + (size_t)i * 32;
        for (int j = 0; j < 32; ++j) ao[j] = row[j] * inv;
    }
}

// ---------------------------------------------------------------------------
// Stage 4a: P[b] = blockdiag(attn[b]) @ w_v     (256x256 per batch)
// ---------------------------------------------------------------------------
__global__ __launch_bounds__(256) void pmat_kernel(const float* __restrict__ attn,
                                                   const float* __restrict__ w_kv,
                                                   float* __restrict__ P) {
    int b = blockIdx.x, tid = threadIdx.x;
    const float* wv = w_kv + 256 * 256;               // bottom half = v-proj
    const float* at = attn + (size_t)b * 8192;        // [8][32][32]
    float* Pb = P + (size_t)b * 65536;
    for (int c = 0; c < 256; ++c) {
        int h = c >> 5, i = c & 31;
        float s = 0.f;
        for (int j = 0; j < 32; ++j)
            s += at[h * 1024 + i * 32 + j] * wv[(h * 32 + j) * 256 + tid];
        Pb[(size_t)c * 256 + tid] = s;
    }
}

// ---------------------------------------------------------------------------
// Stage 4b: W2[b] = w_out @ P[b]                (256x256 per batch)
// ---------------------------------------------------------------------------
__global__ __launch_bounds__(256) void weff_kernel(const float* __restrict__ w_out,
                                                   const float* __restrict__ P,
                                                   float* __restrict__ W2) {
    int b = blockIdx.x, tid = threadIdx.x;
    const float* Pb = P + (size_t)b * 65536;
    float* Wb = W2 + (size_t)b * 65536;
    for (int o = 0; o < 256; ++o) {
        float s = 0.f;
        for (int c = 0; c < 256; ++c) s += w_out[o * 256 + c] * Pb[(size_t)c * 256 + tid];
        Wb[(size_t)o * 256 + tid] = s;
    }
}

// ---------------------------------------------------------------------------
// Stage 5: out[b] = W2[b] @ x[b]  via fp32 WMMA.
// Grid: b(4) x rowblk(4: 64 rows) x colblk(64: 128 cols) = 1024 WGs, 256 thr.
// Wave -> 2x2 subtile block; ping-pong LDS + async global->LDS prefetch.
// ---------------------------------------------------------------------------
__global__ __launch_bounds__(256) void outgemm_kernel(const float* __restrict__ W2,
                                                      const float* __restrict__ x,
                                                      float* __restrict__ out) {
    int bid = blockIdx.x;
    int b   = bid >> 8;
    int rem = bid & 255;
    int rb  = rem >> 6;      // 0..3  (64-row band)
    int cb  = rem & 63;      // 0..63 (128-col band)

    const size_t M = 8192;
    const float* Wb = W2 + (size_t)b * 65536 + (size_t)(rb * 64) * 256;
    const float* Xb = x + (size_t)b * 256 * M + (size_t)cb * 128;
    float* Ob = out + (size_t)b * 256 * M + (size_t)(rb * 64) * M + (size_t)cb * 128;

    __shared__ float ldsW[2][64][36];    // 64 rows x 32 k (+pad)
    __shared__ float ldsX[2][32][132];   // 32 k x 128 cols (+pad)

    int tid  = threadIdx.x;
    int wave = tid >> 5;
    int lane = tid & 31;
    int lrow = lane & 15;
    int khalf = (lane < 16) ? 0 : 2;
    int si2 = wave >> 2;              // 0..1 over 4 row-subtiles
    int sj2 = wave & 3;               // 0..3 over 8 col-subtiles

    v8f acc00 = {}, acc01 = {}, acc10 = {}, acc11 = {};

    auto load_slab = [&](int buf, int kc) {
#pragma unroll
        for (int q = 0; q < 2; ++q) {             // W: 64x32 f32 = 512 float4
            int idx = tid + q * 256;
            int row = idx >> 3;
            int c4  = (idx & 7) << 2;
            async_cp16(&ldsW[buf][row][c4], Wb + (size_t)row * 256 + kc + c4);
        }
#pragma unroll
        for (int q = 0; q < 4; ++q) {             // X: 32x128 f32 = 1024 float4
            int idx = tid + q * 256;
            int row = idx >> 5;
            int c4  = (idx & 31) << 2;
            async_cp16(&ldsX[buf][row][c4], Xb + (size_t)(kc + row) * M + c4);
        }
    };

    load_slab(0, 0);
    wait_async0();
    __syncthreads();

    int ra0 = (si2 * 2) * 16 + lrow;
    int ra1 = ra0 + 16;
    int cb0 = (sj2 * 2) * 16 + lrow;
    int cb1 = cb0 + 16;

    for (int kc = 0; kc < 256; kc += 32) {
        int cur = (kc >> 5) & 1;
        if (kc + 32 < 256) load_slab(cur ^ 1, kc + 32);    // prefetch next slab
#pragma unroll
        for (int kk = 0; kk < 32; kk += 4) {
            v2f a0, a1, b0, b1;
            a0.x = ldsW[cur][ra0][kk + khalf];
            a0.y = ldsW[cur][ra0][kk + khalf + 1];
            a1.x = ldsW[cur][ra1][kk + khalf];
            a1.y = ldsW[cur][ra1][kk + khalf + 1];
            b0.x = ldsX[cur][kk + khalf][cb0];
            b0.y = ldsX[cur][kk + khalf + 1][cb0];
            b1.x = ldsX[cur][kk + khalf][cb1];
            b1.y = ldsX[cur][kk + khalf + 1][cb1];
            acc00 = __builtin_amdgcn_wmma_f32_16x16x4_f32(false, a0, false, b0,
                                                          (short)0, acc00, false, false);
            acc01 = __builtin_amdgcn_wmma_f32_16x16x4_f32(false, a0, false, b1,
                                                          (short)0, acc01, false, false);
            acc10 = __builtin_amdgcn_wmma_f32_16x16x4_f32(false, a1, false, b0,
                                                          (short)0, acc10, false, false);
            acc11 = __builtin_amdgcn_wmma_f32_16x16x4_f32(false, a1, false, b1,
                                                          (short)0, acc11, false, false);
        }
        wait_async0();
        __syncthreads();
    }

    int rowoff = (lane < 16) ? 0 : 8;
    v8f accs[2][2] = {{acc00, acc01}, {acc10, acc11}};
#pragma unroll
    for (int u = 0; u < 2; ++u)
#pragma unroll
        for (int v = 0; v < 2; ++v) {
            float* Op = Ob + (size_t)((si2 * 2 + u) * 16 + rowoff) * M +
                        (sj2 * 2 + v) * 16 + lrow;
#pragma unroll
            for (int r = 0; r < 8; ++r) Op[(size_t)r * M] = accs[u][v][r];
        }
}

// ---------------------------------------------------------------------------
// Launcher.  Inputs: x, y, w_kv, w_q, w_out (all f32).  Output: [4,256,8192] f32.
// Workspace (floats):
//   Cpart : 4 parts * 12 * 65536 = 3145728
//   Call  : 12 * 65536           =  786432
//   attn  : 4*8*32*32            =   32768
//   P     : 4 * 65536            =  262144
//   W2    : 4 * 65536            =  262144
// total ~17.96 MB.
// ---------------------------------------------------------------------------
extern "C" void kernel_launch(void* const* d_in, const int* in_sizes, int n_in,
                              void* d_out, int out_size, void* d_ws, size_t ws_size,
                              hipStream_t stream) {
    const float* x     = (const float*)d_in[0];
    const float* y     = (const float*)d_in[1];
    const float* w_kv  = (const float*)d_in[2];
    const float* w_q   = (const float*)d_in[3];
    const float* w_out = (const float*)d_in[4];
    float* out = (float*)d_out;

    float* ws    = (float*)d_ws;
    float* Cpart = ws;
    float* Call  = Cpart + (size_t)4 * 786432;
    float* attn  = Call + 786432;
    float* P     = attn + 32768;
    float* W2    = P + 262144;

    corr_kernel<<<384, 256, 0, stream>>>(x, y, Cpart);
    reduce_kernel<<<768, 256, 0, stream>>>(Cpart, Call);
    attn_kernel<<<32, 256, 0, stream>>>(Call, w_q, w_kv, attn);
    pmat_kernel<<<4, 256, 0, stream>>>(attn, w_kv, P);
    weff_kernel<<<4, 256, 0, stream>>>(w_out, P, W2);
    outgemm_kernel<<<1024, 256, 0, stream>>>(W2, x, out);
}